// StackPointnetFPModule_23055384444935
// MI455X (gfx1250) — compile-verified
//
#include <hip/hip_runtime.h>
#include <hip/hip_bf16.h>

typedef float v2f __attribute__((ext_vector_type(2)));
typedef float v8f __attribute__((ext_vector_type(8)));

#define BN_EPS   1e-5f
#define DIST_EPS 1e-8f

// ---------------------------------------------------------------------------
// Kernel 0: zero the BN statistics region in workspace
// ---------------------------------------------------------------------------
__global__ void k_zero(float* __restrict__ p, int n) {
  int i = blockIdx.x * blockDim.x + threadIdx.x;
  if (i < n) p[i] = 0.0f;
}

// ---------------------------------------------------------------------------
// Kernel 1: three-NN + inverse-distance interpolation + concat
// One thread per unknown point. Block = 256 threads; Ni % 256 == 0 so every
// block lives entirely in one batch. Known points staged through LDS tiles.
// Writes xbuf[n][0:64] = interp, xbuf[n][64:128] = unknown_feats[n].
// ---------------------------------------------------------------------------
__global__ __launch_bounds__(256) void k_three_nn_interp(
    const float* __restrict__ unknown,        // (B*Ni, 3)
    const float* __restrict__ known,          // (B*Mi, 3)
    const float* __restrict__ unknown_feats,  // (B*Ni, C1)
    const float* __restrict__ known_feats,    // (B*Mi, C2)
    float* __restrict__ xbuf,                 // (B*Ni, 128)
    int Ni, int Mi, int C1, int C2)
{
  __shared__ float skx[256];
  __shared__ float sky[256];
  __shared__ float skz[256];

  const int n     = blockIdx.x * 256 + threadIdx.x;
  const int b     = n / Ni;            // uniform across the block
  const int kbase = b * Mi;

  const float ux = unknown[3 * n + 0];
  const float uy = unknown[3 * n + 1];
  const float uz = unknown[3 * n + 2];

  float d0 = 3.0e38f, d1 = 3.0e38f, d2 = 3.0e38f;
  int   i0 = kbase,   i1 = kbase,    i2 = kbase;

  for (int t = 0; t < Mi; t += 256) {
    const int j = kbase + t + threadIdx.x;
    skx[threadIdx.x] = known[3 * j + 0];
    sky[threadIdx.x] = known[3 * j + 1];
    skz[threadIdx.x] = known[3 * j + 2];
    if (t + 256 < Mi)  // speculative prefetch of the next tile (global_prefetch_b8)
      __builtin_prefetch(&known[3 * (j + 256)], 0, 0);
    __syncthreads();

    #pragma unroll 4
    for (int s = 0; s < 256; ++s) {
      const float dx = ux - skx[s];
      const float dy = uy - sky[s];
      const float dz = uz - skz[s];
      const float d  = dx * dx + dy * dy + dz * dz;
      const int   jj = kbase + t + s;
      if (d < d2) {
        if (d < d1) {
          d2 = d1; i2 = i1;
          if (d < d0) { d1 = d0; i1 = i0; d0 = d; i0 = jj; }
          else        { d1 = d;  i1 = jj; }
        } else        { d2 = d;  i2 = jj; }
      }
    }
    __syncthreads();
  }

  // inverse-distance weights (reference: 1/(sqrt(d2)+eps), normalized)
  const float r0 = 1.0f / (sqrtf(fmaxf(d0, 0.0f)) + DIST_EPS);
  const float r1 = 1.0f / (sqrtf(fmaxf(d1, 0.0f)) + DIST_EPS);
  const float r2 = 1.0f / (sqrtf(fmaxf(d2, 0.0f)) + DIST_EPS);
  const float rs = 1.0f / (r0 + r1 + r2);
  const float w0 = r0 * rs, w1 = r1 * rs, w2 = r2 * rs;

  const float* f0 = known_feats + (long)i0 * C2;
  const float* f1 = known_feats + (long)i1 * C2;
  const float* f2 = known_feats + (long)i2 * C2;
  float* xo = xbuf + (long)n * 128;

  #pragma unroll 8
  for (int c = 0; c < C2; ++c)
    xo[c] = w0 * f0[c] + w1 * f1[c] + w2 * f2[c];

  const float* uf = unknown_feats + (long)n * C1;
  #pragma unroll 8
  for (int c = 0; c < C1; ++c)
    xo[C2 + c] = uf[c];
}

// ---------------------------------------------------------------------------
// Kernel 2: Y = X @ W^T (N x 128) using V_WMMA_F32_16X16X4_F32, fused with
// per-column sum / sum-of-squares accumulation for batch-norm.
// Block = 256 threads = 8 waves; each wave computes a 16-row x 128-col slab.
// A 16x4 f32 frag layout (ISA 7.12.2): lanes 0-15 M=lane K={0,1}, lanes 16-31
// M=lane-16 K={2,3} -> a[v] = X[m][k0 + 2*half + v]. B frag mirrors from W.
// C/D layout: lane holds 8 rows of ONE column -> column stats are lane-local.
// ---------------------------------------------------------------------------
__global__ __launch_bounds__(256) void k_gemm_stats(
    const float* __restrict__ X,    // (N, 128)
    const float* __restrict__ W,    // (128, 128) row-major; Y = X * W^T
    float* __restrict__ Y,          // (N, 128)
    float* __restrict__ gsum,       // (128) global column sums
    float* __restrict__ gsq)        // (128) global column sum-of-squares
{
  __shared__ float ssum[128];
  __shared__ float ssq[128];
  if (threadIdx.x < 128) { ssum[threadIdx.x] = 0.0f; ssq[threadIdx.x] = 0.0f; }
  __syncthreads();

  const int wave = threadIdx.x >> 5;
  const int lane = threadIdx.x & 31;
  const int m    = lane & 15;
  const int half = lane >> 4;
  const int r0   = blockIdx.x * 128 + wave * 16;

  v8f acc[8] = {};

  const float* xrow = X + (long)(r0 + m) * 128 + 2 * half;

  for (int k0 = 0; k0 < 128; k0 += 4) {
    v2f a;
    a.x = xrow[k0 + 0];
    a.y = xrow[k0 + 1];
    #pragma unroll
    for (int nt = 0; nt < 8; ++nt) {
      const float* wrow = W + (long)(nt * 16 + m) * 128 + k0 + 2 * half;
      v2f bf;
      bf.x = wrow[0];
      bf.y = wrow[1];
      acc[nt] = __builtin_amdgcn_wmma_f32_16x16x4_f32(
          /*neg_a=*/false, a, /*neg_b=*/false, bf,
          /*c_mod=*/(short)0, acc[nt], /*reuse_a=*/false, /*reuse_b=*/false);
    }
  }

  #pragma unroll
  for (int nt = 0; nt < 8; ++nt) {
    const int col = nt * 16 + m;
    float ls = 0.0f, lq = 0.0f;
    #pragma unroll
    for (int v = 0; v < 8; ++v) {
      const int  row = r0 + v + 8 * half;
      const float val = acc[nt][v];
      Y[(long)row * 128 + col] = val;
      ls += val;
      lq += val * val;
    }
    atomicAdd(&ssum[col], ls);
    atomicAdd(&ssq[col],  lq);
  }
  __syncthreads();

  if (threadIdx.x < 128) {
    atomicAdd(&gsum[threadIdx.x], ssum[threadIdx.x]);
    atomicAdd(&gsq[threadIdx.x],  ssq[threadIdx.x]);
  }
}

// ---------------------------------------------------------------------------
// Kernel 3: out = relu((y - mean) * rsqrt(var + eps) * g + b), float4 vectorized
// ---------------------------------------------------------------------------
__global__ __launch_bounds__(256) void k_bn_relu(
    const float* __restrict__ Y,
    const float* __restrict__ gsum,
    const float* __restrict__ gsq,
    const float* __restrict__ g,
    const float* __restrict__ bb,
    float* __restrict__ Out,
    int total4, float invN)
{
  const int i = blockIdx.x * 256 + threadIdx.x;
  if (i >= total4) return;

  const float4 y = ((const float4*)Y)[i];
  const int c0 = (i * 4) & 127;

  float yv[4] = { y.x, y.y, y.z, y.w };
  float o[4];
  #pragma unroll
  for (int t = 0; t < 4; ++t) {
    const int   c   = c0 + t;
    const float mu  = gsum[c] * invN;
    const float var = gsq[c] * invN - mu * mu;
    const float v   = (yv[t] - mu) * rsqrtf(var + BN_EPS) * g[c] + bb[c];
    o[t] = fmaxf(v, 0.0f);
  }
  ((float4*)Out)[i] = make_float4(o[0], o[1], o[2], o[3]);
}

// ---------------------------------------------------------------------------
// Host launcher
// ---------------------------------------------------------------------------
extern "C" void kernel_launch(void* const* d_in, const int* in_sizes, int n_in,
                              void* d_out, int out_size, void* d_ws, size_t ws_size,
                              hipStream_t stream) {
  const float* unknown       = (const float*)d_in[0];
  // d_in[1] = unknown_batch_cnt (int64, host-side sizes only)
  const float* known         = (const float*)d_in[2];
  // d_in[3] = known_batch_cnt
  const float* unknown_feats = (const float*)d_in[4];
  const float* known_feats   = (const float*)d_in[5];
  const float* W1            = (const float*)d_in[6];
  const float* g1            = (const float*)d_in[7];
  const float* b1            = (const float*)d_in[8];
  const float* W2            = (const float*)d_in[9];
  const float* g2            = (const float*)d_in[10];
  const float* b2            = (const float*)d_in[11];

  const int B  = in_sizes[1];                 // batch count array length
  const int Ni = in_sizes[0] / (3 * B);
  const int Mi = in_sizes[2] / (3 * B);
  const int N  = B * Ni;                      // 65536 rows
  const int C1 = in_sizes[4] / N;             // 64
  const int C2 = in_sizes[5] / (B * Mi);      // 64

  // Workspace layout: xbuf (N*128 f32) | ybuf (N*128 f32) | stats (512 f32)
  float* xbuf  = (float*)d_ws;
  float* ybuf  = xbuf + (size_t)N * 128;
  float* stats = ybuf + (size_t)N * 128;
  float* sum1 = stats + 0;
  float* sq1  = stats + 128;
  float* sum2 = stats + 256;
  float* sq2  = stats + 384;

  const float invN = 1.0f / (float)N;
  const int total4 = (N * 128) / 4;

  k_zero<<<2, 256, 0, stream>>>(stats, 512);

  k_three_nn_interp<<<N / 256, 256, 0, stream>>>(
      unknown, known, unknown_feats, known_feats, xbuf, Ni, Mi, C1, C2);

  k_gemm_stats<<<N / 128, 256, 0, stream>>>(xbuf, W1, ybuf, sum1, sq1);
  k_bn_relu<<<(total4 + 255) / 256, 256, 0, stream>>>(
      ybuf, sum1, sq1, g1, b1, xbuf, total4, invN);

  k_gemm_stats<<<N / 128, 256, 0, stream>>>(xbuf, W2, ybuf, sum2, sq2);
  k_bn_relu<<<(total4 + 255) / 256, 256, 0, stream>>>(
      ybuf, sum2, sq2, g2, b2, (float*)d_out, total4, invN);
}